// QNet_472446402806
// MI455X (gfx1250) — compile-verified
//
#include <hip/hip_runtime.h>

typedef __attribute__((ext_vector_type(2))) float v2f;
typedef __attribute__((ext_vector_type(8))) float v8f;

#define N_NODES 100000
#define N_EDGES 1600000
#define N_OBS   32
#define N_MID   64
#define N_ACT   8
#define M_TILES (N_NODES / 16)   // 6250, exact

// ---------------- degree / normalization ----------------

__global__ void k_deg_init(float* __restrict__ deg) {
  int i = blockIdx.x * blockDim.x + threadIdx.x;
  if (i < N_NODES) deg[i] = 1.0f;  // self-loop contribution
}

__global__ void k_deg_count(const long long* __restrict__ ei, float* __restrict__ deg) {
  int e = blockIdx.x * blockDim.x + threadIdx.x;
  if (e < N_EDGES) {
    long long d = ei[(long long)N_EDGES + e];  // dst row
    atomicAdd(&deg[d], 1.0f);
  }
}

__global__ void k_dinv(float* __restrict__ deg) {
  int i = blockIdx.x * blockDim.x + threadIdx.x;
  if (i < N_NODES) deg[i] = rsqrtf(deg[i]);  // deg >= 1 always (self-loops)
}

// ---------------- GEMM1: h = x @ W1 (fp32 WMMA 16x16x4) ----------------
// One wave computes one 16x64 output tile.
// A 16x4 fp32 fragment layout: lanes 0-15 hold K={k,k+1}, lanes 16-31 hold K={k+2,k+3}.
// B 4x16 fragment: N = lane&15 striped across lanes, same K split.

__global__ void k_gemm1(const float* __restrict__ x, const float* __restrict__ W1,
                        float* __restrict__ h) {
  int wave = (blockIdx.x * blockDim.x + threadIdx.x) >> 5;
  int lane = threadIdx.x & 31;
  if (wave >= M_TILES) return;  // wave-uniform exit, EXEC stays all-1s

  int mrow  = wave * 16 + (lane & 15);
  int khalf = (lane >> 4) << 1;  // 0 or 2

  v2f a[8];
#pragma unroll
  for (int ks = 0; ks < 8; ++ks) {
    int k = ks * 4 + khalf;
    a[ks].x = x[mrow * N_OBS + k];
    a[ks].y = x[mrow * N_OBS + k + 1];
  }

  int ncol = lane & 15;
#pragma unroll
  for (int nj = 0; nj < 4; ++nj) {
    int n = nj * 16 + ncol;
    v8f c = {0.f, 0.f, 0.f, 0.f, 0.f, 0.f, 0.f, 0.f};
#pragma unroll
    for (int ks = 0; ks < 8; ++ks) {
      int k = ks * 4 + khalf;
      v2f b;
      b.x = W1[k * N_MID + n];
      b.y = W1[(k + 1) * N_MID + n];
      c = __builtin_amdgcn_wmma_f32_16x16x4_f32(false, a[ks], false, b,
                                                (short)0, c, false, false);
    }
    // C/D layout: VGPR r -> M = r (lanes 0-15) or r+8 (lanes 16-31), N = lane&15
#pragma unroll
    for (int r = 0; r < 8; ++r) {
      int row = wave * 16 + r + ((lane >> 4) << 3);
      h[row * N_MID + n] = c[r];
    }
  }
}

// ---------------- agg init: self-loop term agg[i] = h[i] * dinv[i]^2 ----------------

__global__ void k_agg_init(const float* __restrict__ h, const float* __restrict__ dinv,
                           float* __restrict__ agg) {
  int t = blockIdx.x * blockDim.x + threadIdx.x;  // one float4 chunk, N*16 total
  if (t < N_NODES * 16) {
    int i = t >> 4;
    float s = dinv[i];
    float nrm = s * s;
    float4 v = ((const float4*)h)[t];
    float4 o;
    o.x = v.x * nrm; o.y = v.y * nrm; o.z = v.z * nrm; o.w = v.w * nrm;
    ((float4*)agg)[t] = o;
  }
}

// ---------------- edge scatter: agg[dst] += h[src] * dinv[src]*dinv[dst] ----------------

__global__ void k_scatter(const long long* __restrict__ ei, const float* __restrict__ h,
                          const float* __restrict__ dinv, float* __restrict__ agg) {
  int t = blockIdx.x * blockDim.x + threadIdx.x;  // (edge, float4-chunk), E*16 total
  if (t >= N_EDGES * 16) return;
  int e = t >> 4;
  int c = t & 15;
  long long s = ei[e];
  long long d = ei[(long long)N_EDGES + e];
  float nrm = dinv[s] * dinv[d];
  float4 v = ((const float4*)h)[s * 16 + c];
  float* dst = agg + d * (long long)N_MID + c * 4;
  atomicAdd(dst + 0, v.x * nrm);
  atomicAdd(dst + 1, v.y * nrm);
  atomicAdd(dst + 2, v.z * nrm);
  atomicAdd(dst + 3, v.w * nrm);
}

// ---------------- final: out = tanh(agg + b1) @ W2 + b2 (fp32 WMMA) ----------------
// W2 [64x8] zero-padded to N=16 in registers.

__global__ void k_final(const float* __restrict__ agg, const float* __restrict__ b1,
                        const float* __restrict__ W2, const float* __restrict__ b2,
                        float* __restrict__ out) {
  int wave = (blockIdx.x * blockDim.x + threadIdx.x) >> 5;
  int lane = threadIdx.x & 31;
  if (wave >= M_TILES) return;

  int mrow  = wave * 16 + (lane & 15);
  int khalf = (lane >> 4) << 1;

  v2f a[16];
#pragma unroll
  for (int ks = 0; ks < 16; ++ks) {
    int k = ks * 4 + khalf;
    a[ks].x = tanhf(agg[mrow * N_MID + k] + b1[k]);
    a[ks].y = tanhf(agg[mrow * N_MID + k + 1] + b1[k + 1]);
  }

  int n = lane & 15;
  v8f c = {0.f, 0.f, 0.f, 0.f, 0.f, 0.f, 0.f, 0.f};
#pragma unroll
  for (int ks = 0; ks < 16; ++ks) {
    int k = ks * 4 + khalf;
    v2f b;
    b.x = (n < N_ACT) ? W2[k * N_ACT + n] : 0.0f;
    b.y = (n < N_ACT) ? W2[(k + 1) * N_ACT + n] : 0.0f;
    c = __builtin_amdgcn_wmma_f32_16x16x4_f32(false, a[ks], false, b,
                                              (short)0, c, false, false);
  }

  if (n < N_ACT) {
    float bias = b2[n];
#pragma unroll
    for (int r = 0; r < 8; ++r) {
      int row = wave * 16 + r + ((lane >> 4) << 3);
      out[row * N_ACT + n] = c[r] + bias;
    }
  }
}

// ---------------- launch ----------------

extern "C" void kernel_launch(void* const* d_in, const int* in_sizes, int n_in,
                              void* d_out, int out_size, void* d_ws, size_t ws_size,
                              hipStream_t stream) {
  const float*     x  = (const float*)d_in[0];
  const long long* ei = (const long long*)d_in[1];  // int64 per reference
  const float*     W1 = (const float*)d_in[2];
  const float*     b1 = (const float*)d_in[3];
  const float*     W2 = (const float*)d_in[4];
  const float*     b2 = (const float*)d_in[5];
  float*           out = (float*)d_out;

  char* ws = (char*)d_ws;
  size_t off = 0;
  float* deg = (float*)(ws + off);                               // N floats (becomes dinv)
  off += (((size_t)N_NODES * sizeof(float)) + 255) & ~(size_t)255;
  float* h = (float*)(ws + off);                                 // N x 64
  off += (((size_t)N_NODES * N_MID * sizeof(float)) + 255) & ~(size_t)255;
  float* agg = (float*)(ws + off);                               // N x 64
  (void)ws_size; (void)in_sizes; (void)n_in; (void)out_size;

  const int B = 256;
  k_deg_init <<<(N_NODES + B - 1) / B, B, 0, stream>>>(deg);
  k_deg_count<<<(N_EDGES + B - 1) / B, B, 0, stream>>>(ei, deg);
  k_dinv     <<<(N_NODES + B - 1) / B, B, 0, stream>>>(deg);

  int gemm_threads = M_TILES * 32;  // one wave per 16-row tile
  k_gemm1<<<(gemm_threads + B - 1) / B, B, 0, stream>>>(x, W1, h);

  k_agg_init<<<(N_NODES * 16 + B - 1) / B, B, 0, stream>>>(h, deg, agg);
  k_scatter <<<(N_EDGES * 16 + B - 1) / B, B, 0, stream>>>(ei, h, deg, agg);

  k_final<<<(gemm_threads + B - 1) / B, B, 0, stream>>>(agg, b1, W2, b2, out);
}